// TripletLoss_6536940224733
// MI455X (gfx1250) — compile-verified
//
#include <hip/hip_runtime.h>
#include <hip/hip_bf16.h>

typedef __attribute__((ext_vector_type(2))) float v2f;
typedef __attribute__((ext_vector_type(8))) float v8f;

#define N 8192
#define D 128
#define TILE 128
#define LSTR 132   // LDS row stride (floats): 4-bank advance/row -> conflict-free b64 frag loads
#define MARGIN 0.3f

// ---------------------------------------------------------------------------
// Kernel 0: per-row squared norms + init per-row max/min accumulators.
// One wave per row: 32 lanes x float4 = 128 floats.
// ---------------------------------------------------------------------------
__global__ __launch_bounds__(256) void triplet_prep(const float* __restrict__ X,
                                                    float* __restrict__ sq,
                                                    int* __restrict__ gAp,
                                                    int* __restrict__ gAn) {
    const int lane = threadIdx.x & 31;
    const int row  = (blockIdx.x * blockDim.x + threadIdx.x) >> 5;
    if (row >= N) return;
    float4 v = ((const float4*)(X + (size_t)row * D))[lane];
    float s = v.x * v.x + v.y * v.y + v.z * v.z + v.w * v.w;
#pragma unroll
    for (int off = 16; off >= 1; off >>= 1) s += __shfl_xor(s, off, 32);
    if (lane == 0) {
        sq[row]  = s;
        gAp[row] = 0;           // dist >= 0; diagonal guarantees max >= 0
        gAn[row] = 0x7F800000;  // +inf bits; int-min == fp-min for nonneg floats
    }
}

// ---------------------------------------------------------------------------
// Kernel 1: fused  G = X·X^T (fp32 WMMA)  ->  dist  ->  store  ->  masked
// row max/min mining (shuffle + LDS atomics + 1 global atomic per row/tile).
// Block = 256 threads (8 waves, 2x4), tile = 128x128, full K = 128 in LDS.
// Tile staging uses GLOBAL_LOAD_ASYNC_TO_LDS_B128 (ASYNCcnt path, no VGPR
// round-trip); the K loop is software-pipelined one k-step deep.
// ---------------------------------------------------------------------------
__global__ __launch_bounds__(256) void triplet_dist(const float* __restrict__ X,
                                                    const int* __restrict__ tgt,
                                                    const float* __restrict__ sq,
                                                    int* __restrict__ gAp,
                                                    int* __restrict__ gAn,
                                                    float* __restrict__ out) {
    extern __shared__ float smem[];
    float* As    = smem;                    // 128 x 132
    float* Bs    = smem + TILE * LSTR;      // 128 x 132
    float* sqR   = smem + 2 * TILE * LSTR;  // 128
    float* sqC   = sqR + TILE;              // 128
    int*   tgR   = (int*)(sqC + TILE);      // 128
    int*   tgC   = tgR + TILE;              // 128
    int*   redAp = tgC + TILE;              // 128 (int-bit fp)
    int*   redAn = redAp + TILE;            // 128 (int-bit fp)

    const int tid     = threadIdx.x;
    const int rowBase = blockIdx.y * TILE;
    const int colBase = blockIdx.x * TILE;

    // Async-stage A rows and B rows (columns of the output): 128x128 fp32 each,
    // 16B per lane per op, LDS offset = low 32 bits of the generic LDS pointer.
    for (int i = tid; i < TILE * (D / 4); i += 256) {
        const int r  = i >> 5;          // 32 float4 per row
        const int c4 = (i & 31) << 2;
        const unsigned la = (unsigned)(size_t)(As + r * LSTR + c4);
        const unsigned lb = (unsigned)(size_t)(Bs + r * LSTR + c4);
        const float* ga = X + (size_t)(rowBase + r) * D + c4;
        const float* gb = X + (size_t)(colBase + r) * D + c4;
        asm volatile("global_load_async_to_lds_b128 %0, %1, off"
                     :: "v"(la), "v"(ga) : "memory");
        asm volatile("global_load_async_to_lds_b128 %0, %1, off"
                     :: "v"(lb), "v"(gb) : "memory");
    }

    if (tid < TILE) {
        redAp[tid] = 0;
        redAn[tid] = 0x7F800000;
        sqR[tid]   = sq[rowBase + tid];
        tgR[tid]   = tgt[rowBase + tid];
    } else {
        const int c = tid - TILE;
        sqC[c] = sq[colBase + c];
        tgC[c] = tgt[colBase + c];
    }

    asm volatile("s_wait_asynccnt 0" ::: "memory");
    __syncthreads();

    const int lane  = tid & 31;
    const int l15   = lane & 15;
    const int hi    = lane >> 4;       // frag half: K=k,k+1 vs K=k+2,k+3
    const int wave  = tid >> 5;
    const int mBase = (wave >> 2) * 64;  // 2 wave-rows of 64
    const int nBase = (wave & 3) * 32;   // 4 wave-cols of 32

    const float* aRow[4];
    const float* bRow[2];
#pragma unroll
    for (int m = 0; m < 4; ++m) aRow[m] = As + (mBase + m * 16 + l15) * LSTR + 2 * hi;
#pragma unroll
    for (int n = 0; n < 2; ++n) bRow[n] = Bs + (nBase + n * 16 + l15) * LSTR + 2 * hi;

    v8f acc[4][2];
#pragma unroll
    for (int m = 0; m < 4; ++m)
#pragma unroll
        for (int n = 0; n < 2; ++n) acc[m][n] = (v8f){0.f,0.f,0.f,0.f,0.f,0.f,0.f,0.f};

    // Prologue: fragments for k = 0.
    v2f a[4], b[2];
#pragma unroll
    for (int m = 0; m < 4; ++m) a[m] = *(const v2f*)(aRow[m]);
#pragma unroll
    for (int n = 0; n < 2; ++n) b[n] = *(const v2f*)(bRow[n]);

    // K loop, double-buffered: load k+4 frags before issuing k's 8 WMMAs.
#pragma unroll 4
    for (int k = 0; k < D; k += 4) {
        const int k2 = (k + 4) & (D - 1);   // wraps to 0 on last iter (harmless)
        v2f a2[4], b2[2];
#pragma unroll
        for (int m = 0; m < 4; ++m) a2[m] = *(const v2f*)(aRow[m] + k2);
#pragma unroll
        for (int n = 0; n < 2; ++n) b2[n] = *(const v2f*)(bRow[n] + k2);
#pragma unroll
        for (int m = 0; m < 4; ++m)
#pragma unroll
            for (int n = 0; n < 2; ++n)
                acc[m][n] = __builtin_amdgcn_wmma_f32_16x16x4_f32(
                    false, a[m], false, b[n], (short)0, acc[m][n], false, false);
#pragma unroll
        for (int m = 0; m < 4; ++m) a[m] = a2[m];
#pragma unroll
        for (int n = 0; n < 2; ++n) b[n] = b2[n];
    }

    const float INF = __int_as_float(0x7F800000);

    // Epilogue: dist + store + masked row-wise mining.
#pragma unroll
    for (int m = 0; m < 4; ++m) {
#pragma unroll
        for (int t = 0; t < 8; ++t) {
            const int   rLoc = mBase + m * 16 + t + 8 * hi;  // C layout: VGPR t -> M=t / t+8
            const float sqi  = sqR[rLoc];
            const int   ti   = tgR[rLoc];
            float apv = 0.0f;   // masked-out -> 0 (never exceeds true max: diag gives 0)
            float anv = INF;    // masked-out -> +inf
#pragma unroll
            for (int n = 0; n < 2; ++n) {
                const int   cLoc = nBase + n * 16 + l15;
                const float g    = acc[m][n][t];
                float sqd = sqi + sqC[cLoc] - 2.0f * g;
                sqd = fmaxf(sqd, 0.0f);
                const float dv = (sqd > 0.0f) ? sqrtf(sqd) : 0.0f;
                out[(size_t)(rowBase + rLoc) * N + (colBase + cLoc)] = dv;
                const bool same = (ti == tgC[cLoc]);
                apv = fmaxf(apv, same ? dv : 0.0f);
                anv = fminf(anv, same ? INF : dv);
            }
            // reduce across the 16 lanes that share this row
#pragma unroll
            for (int off = 8; off >= 1; off >>= 1) {
                apv = fmaxf(apv, __shfl_xor(apv, off, 16));
                anv = fminf(anv, __shfl_xor(anv, off, 16));
            }
            if (l15 == 0) {
                atomicMax(&redAp[rLoc], __float_as_int(apv));
                atomicMin(&redAn[rLoc], __float_as_int(anv));
            }
        }
    }
    __syncthreads();

    // one global atomic per row per tile
    if (tid < TILE) {
        atomicMax(&gAp[rowBase + tid], redAp[tid]);
    } else {
        atomicMin(&gAn[rowBase + tid - TILE], redAn[tid - TILE]);
    }
}

// ---------------------------------------------------------------------------
// Kernel 2: loss = mean(relu(ap - an + margin)), single block.
// ---------------------------------------------------------------------------
__global__ __launch_bounds__(256) void triplet_loss(const int* __restrict__ gAp,
                                                    const int* __restrict__ gAn,
                                                    float* __restrict__ out) {
    __shared__ float red[8];
    float s = 0.0f;
    for (int i = threadIdx.x; i < N; i += 256) {
        const float ap = __int_as_float(gAp[i]);
        const float an = __int_as_float(gAn[i]);
        s += fmaxf(ap - an + MARGIN, 0.0f);
    }
#pragma unroll
    for (int off = 16; off >= 1; off >>= 1) s += __shfl_xor(s, off, 32);
    if ((threadIdx.x & 31) == 0) red[threadIdx.x >> 5] = s;
    __syncthreads();
    if (threadIdx.x < 8) {
        float v = red[threadIdx.x];
#pragma unroll
        for (int off = 4; off >= 1; off >>= 1) v += __shfl_xor(v, off, 8);
        if (threadIdx.x == 0) out[0] = v / (float)N;
    }
}

// ---------------------------------------------------------------------------
extern "C" void kernel_launch(void* const* d_in, const int* in_sizes, int n_in,
                              void* d_out, int out_size, void* d_ws, size_t ws_size,
                              hipStream_t stream) {
    const float* X   = (const float*)d_in[0];
    const int*   tgt = (const int*)d_in[1];
    float*       out = (float*)d_out;          // out[0]=loss, out[1..]=dist (row-major)

    float* sq  = (float*)d_ws;
    int*   gAp = (int*)d_ws + N;
    int*   gAn = (int*)d_ws + 2 * N;

    triplet_prep<<<(N * 32) / 256, 256, 0, stream>>>(X, sq, gAp, gAn);

    dim3 grid(N / TILE, N / TILE);
    const size_t shmem = (size_t)(2 * TILE * LSTR + 2 * TILE) * sizeof(float)
                       + (size_t)(4 * TILE) * sizeof(int);
    triplet_dist<<<grid, 256, shmem, stream>>>(X, tgt, sq, gAp, gAn, out + 1);

    triplet_loss<<<1, 256, 0, stream>>>(gAp, gAn, out);
}